// PointCloudPatchEmbedding_12816182411336
// MI455X (gfx1250) — compile-verified
//
#include <hip/hip_runtime.h>
#include <hip/hip_bf16.h>
#include <math.h>

#define RESV   64
#define R3     (RESV*RESV*RESV)      // 262144 voxels per batch
#define PNTS   262144                // points per batch (2^18)
#define NBATCH 16
#define HIDDEN 384
#define KTOT   192                   // 3 * 4*4*4
#define TOKB   4096                  // 16^3 tokens per batch
#define NTOK   (NBATCH*TOKB)         // 65536 tokens total

typedef __attribute__((ext_vector_type(2))) float v2f;
typedef __attribute__((ext_vector_type(8))) float v8f;

// workspace (floats): [0..47]=mean(b,c) [48..63]=maxdist(b)
// [64 ..) grid4: NBATCH * R3 * 4  (sumx,sumy,sumz,count)  = 64 MB
#define WS_GRID_OFF 64
#define GRID_FLOATS ((size_t)NBATCH * R3 * 4)

__global__ void k_zero(float* ws, size_t n) {
  size_t i = (size_t)blockIdx.x * blockDim.x + threadIdx.x;
  size_t stride = (size_t)gridDim.x * blockDim.x;
  for (; i < n; i += stride) ws[i] = 0.f;
}

__global__ void k_mean(const float* __restrict__ pc, float* __restrict__ ws) {
  int bc = blockIdx.x;                      // b*3 + c, 0..47
  const float* src = pc + (size_t)bc * PNTS;
  float s = 0.f;
  for (int p = threadIdx.x; p < PNTS; p += blockDim.x) s += src[p];
  __shared__ float red[256];
  red[threadIdx.x] = s; __syncthreads();
  for (int off = 128; off > 0; off >>= 1) {
    if (threadIdx.x < off) red[threadIdx.x] += red[threadIdx.x + off];
    __syncthreads();
  }
  if (threadIdx.x == 0) ws[bc] = red[0] * (1.f / PNTS);
}

__global__ void k_maxdist(const float* __restrict__ pc, float* __restrict__ ws) {
  int b = blockIdx.x;                       // 0..15
  float m0 = ws[b*3+0], m1 = ws[b*3+1], m2 = ws[b*3+2];
  const float* x = pc + ((size_t)b*3 + 0) * PNTS;
  const float* y = pc + ((size_t)b*3 + 1) * PNTS;
  const float* z = pc + ((size_t)b*3 + 2) * PNTS;
  float mx = 0.f;
  for (int p = threadIdx.x; p < PNTS; p += blockDim.x) {
    float dx = x[p]-m0, dy = y[p]-m1, dz = z[p]-m2;
    mx = fmaxf(mx, sqrtf(dx*dx + dy*dy + dz*dz));
  }
  __shared__ float red[256];
  red[threadIdx.x] = mx; __syncthreads();
  for (int off = 128; off > 0; off >>= 1) {
    if (threadIdx.x < off) red[threadIdx.x] = fmaxf(red[threadIdx.x], red[threadIdx.x+off]);
    __syncthreads();
  }
  if (threadIdx.x == 0) ws[48 + b] = red[0];
}

__global__ void k_voxelize(const float* __restrict__ pc, float* __restrict__ ws) {
  size_t idx = (size_t)blockIdx.x * blockDim.x + threadIdx.x;
  if (idx >= (size_t)NBATCH * PNTS) return;
  int b = (int)(idx >> 18);
  int p = (int)(idx & (PNTS - 1));
  float* grid = ws + WS_GRID_OFF;
  float inv = 1.f / (2.f * ws[48 + b]);
  int   vi[3]; float val[3];
#pragma unroll
  for (int c = 0; c < 3; ++c) {
    float v  = pc[((size_t)(b*3 + c) << 18) + p];
    val[c]   = v;
    float nc = (v - ws[b*3 + c]) * inv + 0.5f;
    float f  = fminf(fmaxf(nc * (float)RESV, 0.f), (float)(RESV - 1));
    vi[c]    = (int)rintf(f);                    // round-half-even == jnp.round
  }
  int lin = (vi[0]*RESV + vi[1])*RESV + vi[2];
  float* cell = grid + (((size_t)b * R3 + lin) << 2);
  atomicAdd(cell + 0, val[0]);
  atomicAdd(cell + 1, val[1]);
  atomicAdd(cell + 2, val[2]);
  atomicAdd(cell + 3, 1.f);
}

// Block: 256 thr (8 waves). Computes 16 tokens x 128 hidden.
// Wave w -> n-subtile [nBlock*128 + w*16, +16). 48 x V_WMMA_F32_16X16X4_F32 per wave.
__global__ void __launch_bounds__(256)
k_patch_gemm(const float* __restrict__ ws,
             const float* __restrict__ w,
             const float* __restrict__ bias,
             float* __restrict__ out) {
  const float* grid = ws + WS_GRID_OFF;
  __shared__ __align__(16) float ldsA[16 * 196];   // 196 pitch -> conflict-free ds reads

  int tokenBase = blockIdx.x * 16;
  int nBlock    = blockIdx.y;
  int tid       = threadIdx.x;

  // Cooperative A-tile build: 16 tokens x 192 patch values, fused avg = sum/count.
  for (int e = tid; e < 16 * KTOT; e += blockDim.x) {
    int m = e / KTOT, k = e - m * KTOT;
    int tok = tokenBase + m;
    int b = tok >> 12;                 // / 4096
    int t = tok & 4095;
    int d = t >> 8, hh = (t >> 4) & 15, wv = t & 15;
    int c = k >> 6, rem = k & 63;
    int kd = rem >> 4, kh = (rem >> 2) & 3, kw = rem & 3;
    int vox = ((d*4 + kd)*RESV + (hh*4 + kh))*RESV + (wv*4 + kw);
    const float* cell = grid + (((size_t)b * R3 + vox) << 2);
    float cnt = cell[3];
    ldsA[m * 196 + k] = (cnt > 0.f) ? (cell[c] / cnt) : 0.f;
  }
  __syncthreads();

  int wave = tid >> 5, lane = tid & 31;
  int half = lane >> 4;                // k-pair select per ISA A/B layout
  int ln   = lane & 15;                // m for A frag, n for B/C/D frags
  int nGlob = nBlock * 128 + wave * 16 + ln;
  const float* wrow = w + (size_t)nGlob * KTOT;   // conv_w[n][k] row-major

  v8f acc = {};
#pragma unroll
  for (int k = 0; k < KTOT; k += 4) {
    int k0 = k + half * 2;
    v2f a  = *(const v2f*)(&ldsA[ln * 196 + k0]);   // A[m][k0], A[m][k0+1]
    v2f bb = *(const v2f*)(wrow + k0);              // B[k0][n], B[k0+1][n]
    acc = __builtin_amdgcn_wmma_f32_16x16x4_f32(
        /*neg_a=*/false, a, /*neg_b=*/false, bb,
        /*c_mod=*/(short)0, acc, /*reuse_a=*/false, /*reuse_b=*/false);
  }

  float bv = bias[nGlob];
#pragma unroll
  for (int r = 0; r < 8; ++r) {
    int row = tokenBase + r + 8 * half;             // C/D layout: m = r + 8*half
    out[(size_t)row * HIDDEN + nGlob] = acc[r] + bv;
  }
}

extern "C" void kernel_launch(void* const* d_in, const int* in_sizes, int n_in,
                              void* d_out, int out_size, void* d_ws, size_t ws_size,
                              hipStream_t stream) {
  const float* pc = (const float*)d_in[0];   // [16,3,262144]
  const float* cw = (const float*)d_in[1];   // [384,3,4,4,4]
  const float* cb = (const float*)d_in[2];   // [384]
  float* out = (float*)d_out;                // [16,4096,384]
  float* ws  = (float*)d_ws;

  size_t nzero = WS_GRID_OFF + GRID_FLOATS;
  k_zero<<<8192, 256, 0, stream>>>(ws, nzero);
  k_mean<<<NBATCH * 3, 256, 0, stream>>>(pc, ws);
  k_maxdist<<<NBATCH, 256, 0, stream>>>(pc, ws);

  int npts = NBATCH * PNTS;
  k_voxelize<<<(npts + 255) / 256, 256, 0, stream>>>(pc, ws);

  dim3 g(NTOK / 16, HIDDEN / 128);
  k_patch_gemm<<<g, 256, 0, stream>>>(ws, cw, cb, out);
}